// ChamferLoss_32856499815265
// MI455X (gfx1250) — compile-verified
//
#include <hip/hip_runtime.h>
#include <hip/hip_bf16.h>

typedef float v2f __attribute__((ext_vector_type(2)));
typedef float v8f __attribute__((ext_vector_type(8)));

#define NPTS   2048
#define NTILES (NPTS / 16)     // 128
#define BATCH  32
#define WAVES_PER_BLOCK 8
#define BLOCKS_PER_BATCH (NTILES / WAVES_PER_BLOCK)   // 16

// Math:  A = [-2x0, -2x1, -2x2, 1]   (per-wave, fixed)
//        B = [ y0,   y1,   y2, ||y||^2 ]
//   =>   D[i][j] = ||y_j||^2 - 2<x_i, y_j>
//   =>   min_j dist[i][j] = ||x_i||^2 + min_j D[i][j]   (x-norm hoisted out)
//
// All 8 waves of a block share one batch; the block cooperatively prepacks
// the whole target set into LDS in the exact WMMA B-operand register layout:
//   tile t, lane L (<16):  [y0, y1]  of point t*16+L     (K=0,1 rows)
//   tile t, lane L (>=16): [y2, yn]  of point t*16+L-16  (K=2,3 rows)
// Hot loop per tile: ds_load_b64 + v_wmma_f32_16x16x4_f32 + fused min3s.
__global__ __launch_bounds__(256) void chamfer_dir_kernel(
    const float* __restrict__ P,      // [BATCH, NPTS, 3] query points  (A)
    const float* __restrict__ Q,      // [BATCH, NPTS, 3] target points (B)
    float* __restrict__ partial)      // [BATCH * NTILES]
{
    __shared__ float sB[NTILES * 32 * 2];   // 32 KB: per-lane B operands

    const unsigned lane   = threadIdx.x & 31u;
    const unsigned warp   = threadIdx.x >> 5;
    const unsigned b      = blockIdx.x / BLOCKS_PER_BATCH;
    const unsigned tile_i = (blockIdx.x % BLOCKS_PER_BATCH) * WAVES_PER_BLOCK + warp;
    const unsigned hi     = lane >> 4;     // 0: holds K=0,1   1: holds K=2,3
    const unsigned sub    = lane & 15u;

    const float* __restrict__ pb = P + (size_t)b * NPTS * 3;
    const float* __restrict__ qb = Q + (size_t)b * NPTS * 3;

    // ---- cooperative prepack of all target points into LDS (B layout)
    for (unsigned p = threadIdx.x; p < NPTS; p += blockDim.x) {
        const float* yq = qb + (size_t)p * 3u;
        const float y0 = yq[0], y1 = yq[1], y2 = yq[2];
        const float yn = y0 * y0 + y1 * y1 + y2 * y2;
        const unsigned t = p >> 4, s = p & 15u;
        v2f lo; lo.x = y0; lo.y = y1;          // lanes 0-15: K=0,1
        v2f hiv; hiv.x = y2; hiv.y = yn;       // lanes 16-31: K=2,3 (row3 = yn)
        *(v2f*)&sB[(t * 32u + s) * 2u]       = lo;
        *(v2f*)&sB[(t * 32u + 16u + s) * 2u] = hiv;
    }

    // ---- A-matrix (fixed for the whole wave): [-2x0, -2x1, -2x2, 1]
    const float* xp = pb + (size_t)(tile_i * 16u + sub) * 3u;
    const float x0 = xp[0], x1 = xp[1], x2 = xp[2];
    const float xn = x0 * x0 + x1 * x1 + x2 * x2;
    v2f A;
    A.x = hi ? (-2.0f * x2) : (-2.0f * x0);   // K=2 : K=0
    A.y = hi ? 1.0f         : (-2.0f * x1);   // K=3 : K=1

    // Broadcast query norms to match D-tile layout: i = v + 8*hi.
    float xnv[8];
#pragma unroll
    for (int v = 0; v < 8; ++v)
        xnv[v] = __shfl(xn, v + 8 * (int)hi, 32);

    float mins[8];
#pragma unroll
    for (int v = 0; v < 8; ++v) mins[v] = 3.402823466e38f;

    __syncthreads();

    // ---- hot loop: ds_load_b64 + WMMA + min
    const float* sBlane = &sB[(size_t)lane * 2u];
    for (int t = 0; t < NTILES; ++t) {
        const v2f Bm = *(const v2f*)&sBlane[(size_t)t * 64u];
        v8f c = {};
        // D[i][j] = ||y_j||^2 - 2 <x_i, y_j>
        v8f dot = __builtin_amdgcn_wmma_f32_16x16x4_f32(
            false, A, false, Bm, (short)0, c, false, false);
#pragma unroll
        for (int v = 0; v < 8; ++v)
            mins[v] = fminf(mins[v], dot[v]);
    }

    // ---- add hoisted ||x_i||^2, then reduce min over j (16-lane groups)
    float s = 0.0f;
#pragma unroll
    for (int v = 0; v < 8; ++v) {
        float m = mins[v] + xnv[v];
        m = fminf(m, __shfl_xor(m, 1, 32));
        m = fminf(m, __shfl_xor(m, 2, 32));
        m = fminf(m, __shfl_xor(m, 4, 32));
        m = fminf(m, __shfl_xor(m, 8, 32));
        s += m;   // lanes 0-15: sum over i=0..7 ; lanes 16-31: i=8..15
    }
    s += __shfl_xor(s, 16, 32);   // full sum of the 16 row-mins

    if (lane == 0) partial[b * NTILES + tile_i] = s;   // deterministic
}

// 1 block, 32 threads: thread b combines the two directed means for batch b,
// takes the max, then a wave-wide mean over batches.
__global__ void chamfer_finalize_kernel(const float* __restrict__ partial,
                                        float* __restrict__ out)
{
    const int b = threadIdx.x;   // 0..31
    float rs = 0.0f, cs = 0.0f;
    for (int t = 0; t < NTILES; ++t) {
        rs += partial[b * NTILES + t];                   // x->y direction
        cs += partial[BATCH * NTILES + b * NTILES + t];  // y->x direction
    }
    float v = fmaxf(rs * (1.0f / (float)NPTS), cs * (1.0f / (float)NPTS));
    v += __shfl_xor(v, 1, 32);
    v += __shfl_xor(v, 2, 32);
    v += __shfl_xor(v, 4, 32);
    v += __shfl_xor(v, 8, 32);
    v += __shfl_xor(v, 16, 32);
    if (b == 0) out[0] = v * (1.0f / (float)BATCH);
}

extern "C" void kernel_launch(void* const* d_in, const int* in_sizes, int n_in,
                              void* d_out, int out_size, void* d_ws, size_t ws_size,
                              hipStream_t stream) {
    const float* x = (const float*)d_in[0];   // [32, 2048, 3]
    const float* y = (const float*)d_in[1];   // [32, 2048, 3]
    float* out = (float*)d_out;               // scalar
    float* ws  = (float*)d_ws;                // 2 * 32 * 128 floats = 32 KB

    const dim3 block(256);
    const dim3 grid(BATCH * BLOCKS_PER_BATCH);   // 512 blocks

    // Direction 1: for each x point, min over y  -> "row" term
    chamfer_dir_kernel<<<grid, block, 0, stream>>>(x, y, ws);
    // Direction 2: for each y point, min over x  -> "col" term
    chamfer_dir_kernel<<<grid, block, 0, stream>>>(y, x, ws + BATCH * NTILES);

    chamfer_finalize_kernel<<<1, 32, 0, stream>>>(ws, out);
}